// nu_grid_sample_1468878815897
// MI455X (gfx1250) — compile-verified
//
#include <hip/hip_runtime.h>

// Non-uniform 3x3 Gaussian grid-sample, MI455X (gfx1250, wave32).
// x: (4, 64, 512, 512) f32, coords: (4, 32768, 2) f32 -> out: (4, 64, 32768) f32.
//
// Roofline: ~300 MB HBM traffic -> ~13 us floor @ 23.3 TB/s; real limiter is
// L2 request throughput on scattered 3x3 patches. Design:
//  * one lane per point; tap weights computed once, remapped onto 3 CONTIGUOUS
//    pixels per axis so each patch row is a single global_load_b96
//    (3 L2 requests per 9 taps instead of 9),
//  * channel loop: wave-uniform (scalar) plane base + per-lane 32-bit offset,
//    fully coalesced output stores,
//  * coords tile (2 KB, contiguous) staged into LDS with the CDNA5 Tensor Data
//    Mover (tensor_load_to_lds + s_wait_tensorcnt) - the one place in this
//    gather-bound kernel where a bulk async-DMA path genuinely applies.
// WMMA is deliberately NOT used: the 9-tap contraction is diagonal in n on
// both operands, so any WMMA mapping is block-diagonal (15/16 wasted MACs) and
// would replace b96 row loads with scattered per-k-slot loads, increasing L2
// requests 2-4x on the actual bottleneck.

namespace {
constexpr int kNX = 512;
constexpr int kNY = 512;
constexpr int kPlane = kNX * kNY;
constexpr int kC = 64;
constexpr int kNP = 32768;
constexpr int kB = 4;
constexpr int kThreads = 256;
}  // namespace

// 3-float vector with 4-byte alignment so loads lower to global_load_b96 at
// dword alignment (default ext_vector_type(3) would claim 16B align).
typedef float v3f __attribute__((ext_vector_type(3), aligned(4)));

typedef unsigned int u32x4 __attribute__((ext_vector_type(4)));
typedef int i32x8 __attribute__((ext_vector_type(8)));
typedef int i32x4 __attribute__((ext_vector_type(4)));

#if defined(__has_builtin)
#if __has_builtin(__builtin_amdgcn_tensor_load_to_lds) && \
    __has_builtin(__builtin_amdgcn_s_wait_tensorcnt)
#define NUGS_USE_TDM 1
#endif
#endif
#ifndef NUGS_USE_TDM
#define NUGS_USE_TDM 0
#endif

__device__ __forceinline__ float clampf(float v, float lo, float hi) {
  return fminf(fmaxf(v, lo), hi);
}

__global__ __launch_bounds__(kThreads) void nugs_gather_kernel(
    const float* __restrict__ x, const float* __restrict__ coords,
    float* __restrict__ out) {
  const int gid = blockIdx.x * kThreads + threadIdx.x;  // 0 .. 4*32768-1
  const int b = gid >> 15;          // 32768 points per batch, 128 blocks/batch
  const int n = gid & (kNP - 1);

  __shared__ float2 s_coords[kThreads];  // 2 KB staged coords tile

#if NUGS_USE_TDM
  // --- TDM staging: one wave DMAs the block's contiguous coords tile to LDS.
  // D# per CDNA5 ISA (group0 128b, group1 256b, 1-D tile):
  //   group0: count=1 | lds_addr | global_addr[56:0] | type=2
  //   group1: data_size=2 (4B) | tensor_dim0=512 | tensor_dim1=1 |
  //           tile_dim0=512 (dwords) | tile_dim1=0 (1-D) | dim0_stride=512
  if (threadIdx.x < 32) {  // wave 0 only; TDM ignores EXEC, issues once/wave
    const int n0 = n & ~(kThreads - 1);
    const float* gsrc = coords + (((size_t)b * kNP + n0) << 1);
    const unsigned long long ga = (unsigned long long)(size_t)gsrc;
    // Generic->LDS: hardware uses addr[31:0] as the LDS byte address.
    const unsigned int lds_addr = (unsigned int)(size_t)(void*)s_coords;

    const u32x4 g0 = {
        1u,                                     // count=1, user descriptor
        lds_addr,                               // lds_addr [63:32]
        (unsigned int)(ga & 0xffffffffu),       // global_addr [95:64]
        (unsigned int)((ga >> 32) & 0x01ffffffu) | (2u << 30)  // addr|type=2
    };
    const i32x8 g1 = {
        (int)(2u << 16),    // workgroup_mask=0, data_size=2 (4 bytes)
        (int)(512u << 16),  // atomic_barrier_addr=0 | tensor_dim0.lo16=512
        (int)(1u << 16),    // tensor_dim0.hi16=0    | tensor_dim1.lo16=1
        (int)(512u << 16),  // tensor_dim1.hi16=0    | tile_dim0=512
        0,                  // tile_dim1=0 (1-D), tile_dim2=0
        512,                // tensor_dim0_stride[31:0]=512
        0, 0                // stride hi, tensor_dim1_stride=0
    };
    const i32x4 gz4 = {0, 0, 0, 0};          // groups 2/3 unused (<= 2-D)
    const i32x8 gz8 = {0, 0, 0, 0, 0, 0, 0, 0};
    // clang-23 / therock-10.0 6-arg form (probe-confirmed arity).
    __builtin_amdgcn_tensor_load_to_lds(g0, g1, gz4, gz4, gz8, 0);
    __builtin_amdgcn_s_wait_tensorcnt(0);  // TENSORcnt==0: tile is in LDS
  }
  __syncthreads();
  const float2 cc = s_coords[threadIdx.x];
#else
  const float2 cc = reinterpret_cast<const float2*>(coords)[b * kNP + n];
#endif

  // coords[b, n, 0] -> fast axis (y), coords[b, n, 1] -> slow axis (x)
  const float posy = cc.x * (float)(kNY - 1);
  const float posx = cc.y * (float)(kNX - 1);
  const float rpx = rintf(posx);  // round-half-even, matches jnp.round
  const float rpy = rintf(posy);

  // 9 sub-pixel Gaussian samples per axis, grouped 3-at-a-time.
  // _normal's 1/(s*sqrt(2pi)) constants cancel in the normalization -> drop.
  // exponent: -0.5*((d)/0.5)^2 = -2*d^2.  Reference clips BOTH axes to
  // [0, nx=512] (its quirk) - reproduced here.
  float gx[3] = {0.f, 0.f, 0.f};
  float gy[3] = {0.f, 0.f, 0.f};
#pragma unroll
  for (int r = 0; r < 9; ++r) {
    const float off = 1.5f - 0.375f * (float)r;  // linspace(1.5,-1.5,9)
    const float dx = clampf(rpx - off, 0.f, 512.f) - posx;
    const float dy = clampf(rpy - off, 0.f, 512.f) - posy;
    gx[r / 3] += __expf(-2.f * dx * dx);
    gy[r / 3] += __expf(-2.f * dy * dy);
  }
  const float inv =
      1.f / ((gx[0] + gx[1] + gx[2]) * (gy[0] + gy[1] + gy[2]));

  // Reference taps: ix[i] = clip(rpx + 1 - i, 0, 511), i = 0..2 (same for y).
  // Clipping can duplicate edge pixels; fold the 3 tap weights onto the 3
  // CONTIGUOUS pixels [ix0, ix0+2], ix0 = clip(rpx-1, 0, 509), so each patch
  // row is one b96 load regardless of edge clamping.
  const float ix0f = clampf(rpx - 1.f, 0.f, 509.f);
  const float iy0f = clampf(rpy - 1.f, 0.f, 509.f);
  float wc0 = 0.f, wc1 = 0.f, wc2 = 0.f;  // x (slow axis / row) weights
  float wr0 = 0.f, wr1 = 0.f, wr2 = 0.f;  // y (fast axis / col) weights
#pragma unroll
  for (int i = 0; i < 3; ++i) {
    const int px = (int)(clampf(rpx + 1.f - (float)i, 0.f, 511.f) - ix0f);
    const int py = (int)(clampf(rpy + 1.f - (float)i, 0.f, 511.f) - iy0f);
    wc0 += (px == 0) ? gx[i] : 0.f;
    wc1 += (px == 1) ? gx[i] : 0.f;
    wc2 += (px == 2) ? gx[i] : 0.f;
    wr0 += (py == 0) ? gy[i] : 0.f;
    wr1 += (py == 1) ? gy[i] : 0.f;
    wr2 += (py == 2) ? gy[i] : 0.f;
  }
  wc0 *= inv;
  wc1 *= inv;
  wc2 *= inv;
  const float w00 = wc0 * wr0, w01 = wc0 * wr1, w02 = wc0 * wr2;
  const float w10 = wc1 * wr0, w11 = wc1 * wr1, w12 = wc1 * wr2;
  const float w20 = wc2 * wr0, w21 = wc2 * wr1, w22 = wc2 * wr2;

  const int pix = (int)ix0f * kNY + (int)iy0f;  // per-lane pixel offset
  const float* __restrict__ xb = x + (size_t)b * ((size_t)kC * kPlane);
  float* __restrict__ ob = out + ((size_t)b * kC) * (size_t)kNP + n;

  // Channel loop: plane base is wave-uniform (scalar base + vgpr offset);
  // stores are fully coalesced (consecutive n across the wave).
#pragma unroll 4
  for (int c = 0; c < kC; ++c) {
    const float* __restrict__ plane = xb + (size_t)c * kPlane;
    const v3f r0 = *reinterpret_cast<const v3f*>(plane + pix);
    const v3f r1 = *reinterpret_cast<const v3f*>(plane + pix + kNY);
    const v3f r2 = *reinterpret_cast<const v3f*>(plane + pix + 2 * kNY);
    float acc = w00 * r0.x;
    acc = fmaf(w01, r0.y, acc);
    acc = fmaf(w02, r0.z, acc);
    acc = fmaf(w10, r1.x, acc);
    acc = fmaf(w11, r1.y, acc);
    acc = fmaf(w12, r1.z, acc);
    acc = fmaf(w20, r2.x, acc);
    acc = fmaf(w21, r2.y, acc);
    acc = fmaf(w22, r2.z, acc);
    ob[(size_t)c * kNP] = acc;
  }
}

extern "C" void kernel_launch(void* const* d_in, const int* in_sizes, int n_in,
                              void* d_out, int out_size, void* d_ws,
                              size_t ws_size, hipStream_t stream) {
  (void)in_sizes;
  (void)n_in;
  (void)out_size;
  (void)d_ws;
  (void)ws_size;
  const float* x = reinterpret_cast<const float*>(d_in[0]);
  const float* coords = reinterpret_cast<const float*>(d_in[1]);
  float* out = reinterpret_cast<float*>(d_out);

  const int total = kB * kNP;  // one lane per (b, n)
  dim3 grid(total / kThreads);
  dim3 block(kThreads);
  hipLaunchKernelGGL(nugs_gather_kernel, grid, block, 0, stream, x, coords,
                     out);
}